// Raindrop_56796647522440
// MI455X (gfx1250) — compile-verified
//
#include <hip/hip_runtime.h>
#include <math.h>

// ---------------------------------------------------------------------------
// MI455X (gfx1250) implementation notes:
//  - wave32 only; WMMA 16x16 tiles. We use V_WMMA_F32_16X16X4_F32 (full fp32
//    precision; the model is memory-bound at 23.3 TB/s, so the f32 matrix pipe
//    is more than sufficient and avoids any precision loss).
//  - K is a compile-time template parameter for every GEMM in the model, so
//    the MMA inner loops are fully unrolled straight-line WMMA streams with
//    batched ds_load_2addr_b64 operand fetches (verified in round-2 disasm).
//  - MHA LDS tiles are padded to 20 columns (zero-filled) so all inner-loop
//    LDS reads are unconditional: no s_and_saveexec exec-mask churn.
//  - Fragment layouts per CDNA5 ISA:
//      A (16x4 f32): lane = M (l&15), VGPR0 = K0(lo half)/K2(hi half), VGPR1 = K1/K3
//      B (4x16 f32): lane = N (l&15), VGPR0 = K0/K2, VGPR1 = K1/K3
//      C/D (16x16):  lane = N (l&15), VGPR j = row (j + 8*(l>>4))
// ---------------------------------------------------------------------------

typedef float v2f __attribute__((ext_vector_type(2)));
typedef float v8f __attribute__((ext_vector_type(8)));

#define T_LEN 215
#define T_PAD 224
#define BATCH 512
#define TBROWS (T_LEN * BATCH)

__device__ __forceinline__ v8f wmma4(v2f a, v2f b, v8f c) {
  return __builtin_amdgcn_wmma_f32_16x16x4_f32(false, a, false, b, (short)0, c,
                                               false, false);
}

// ---------------------------------------------------------------------------
// Generic fp32 WMMA GEMM:  C[m, ccol+n] = act( (A[MxK] @ W[KxN] + bias) * scale )
// W read transposed when transB!=0 (W logical [N x K] with row stride ldw).
// Block = 4 waves; each wave owns one 16x16 output tile. A and W staged in LDS
// in 64-deep K chunks (zero-padded, so MMA loops need no per-element guards).
// ---------------------------------------------------------------------------
__device__ __forceinline__ void gemm_stage(const float* __restrict__ A, int lda,
                                           const float* __restrict__ W, int ldw,
                                           int transB, int M, int N, int K,
                                           int kc, int nt, int mt, int wv,
                                           float wt[64][16],
                                           float at[4][16][65]) {
  __syncthreads();
  // stage W chunk (shared by all 4 waves: same n-tile)
#pragma unroll
  for (int i = 0; i < 8; ++i) {
    int idx = threadIdx.x + i * 128;
    int kk = idx >> 4, n = idx & 15;
    int kg = kc + kk, ng = nt + n;
    float v = 0.f;
    if (kg < K && ng < N)
      v = transB ? W[(size_t)ng * ldw + kg] : W[(size_t)kg * ldw + ng];
    wt[kk][n] = v;
  }
  // stage A chunk (per-wave, coalesced along K)
  const int lane = threadIdx.x & 31;
#pragma unroll
  for (int i = 0; i < 32; ++i) {
    int idx = lane + i * 32;
    int m = idx >> 6, kk = idx & 63;
    int mg = mt + m, kg = kc + kk;
    at[wv][m][kk] = (mg < M && kg < K) ? A[(size_t)mg * lda + kg] : 0.f;
  }
  // speculative prefetch of the next A chunk (global_prefetch_b8)
  int pm = mt + (lane & 15);
  if (kc + 64 < K && pm < M)
    __builtin_prefetch(&A[(size_t)pm * lda + kc + 64], 0, 3);
  __syncthreads();
}

template <int KLIM>
__device__ __forceinline__ void gemm_mma(float wt[64][16], float at[16][65],
                                         int half, int l15, v8f& acc) {
#pragma unroll
  for (int k0 = 0; k0 < KLIM; k0 += 4) {
    int ka = k0 + 2 * half;
    v2f a, b;
    a.x = at[l15][ka];
    a.y = at[l15][ka + 1];
    b.x = wt[ka][l15];
    b.y = wt[ka + 1][l15];
    acc = wmma4(a, b, acc);
  }
}

template <int K>
__global__ __launch_bounds__(128) void gemm_wmma_kernel(
    const float* __restrict__ A, int lda, const float* __restrict__ W, int ldw,
    int transB, const float* __restrict__ bias, float* __restrict__ C, int ldc,
    int ccol, int M, int N, float scale, int relu) {
  __shared__ float wt[64][16];
  __shared__ float at[4][16][65];

  const int lane = threadIdx.x & 31;
  const int wv = threadIdx.x >> 5;
  const int half = lane >> 4;
  const int l15 = lane & 15;
  const int nt = blockIdx.x * 16;
  const int mt = (blockIdx.y * 4 + wv) * 16;

  v8f acc{};
  constexpr int KFULL = (K / 64) * 64;
  constexpr int KTAIL4 = ((K - KFULL) + 3) & ~3;  // tail rounded to mult of 4

  for (int kc = 0; kc < KFULL; kc += 64) {
    gemm_stage(A, lda, W, ldw, transB, M, N, K, kc, nt, mt, wv, wt, at);
    gemm_mma<64>(wt, at[wv], half, l15, acc);  // 16 straight-line WMMAs
  }
  if constexpr (KTAIL4 > 0) {
    gemm_stage(A, lda, W, ldw, transB, M, N, K, KFULL, nt, mt, wv, wt, at);
    gemm_mma<KTAIL4>(wt, at[wv], half, l15, acc);
  }

  const int ng = nt + l15;
  if (ng < N) {
    const float bv = bias ? bias[ng] : 0.f;
#pragma unroll
    for (int j = 0; j < 8; ++j) {
      int mg = mt + j + 8 * half;
      if (mg < M) {
        float v = (acc[j] + bv) * scale;
        if (relu) v = fmaxf(v, 0.f);
        C[(size_t)mg * ldc + ccol + ng] = v;
      }
    }
  }
}

// ---------------------------------------------------------------------------
// Positional encoding: h[tb, 36+i] = sin(st), h[tb, 54+i] = cos(st),
// st = times[tb] / 215^(i/17),  i in [0,18)
// ---------------------------------------------------------------------------
__global__ void pe_kernel(const float* __restrict__ times,
                          float* __restrict__ h) {
  int idx = blockIdx.x * blockDim.x + threadIdx.x;
  if (idx >= TBROWS * 18) return;
  int tb = idx / 18;
  int i = idx - tb * 18;
  float ts = powf(215.0f, (float)i * (1.0f / 17.0f));
  float st = times[tb] / ts;
  h[(size_t)tb * 72 + 36 + i] = sinf(st);
  h[(size_t)tb * 72 + 54 + i] = cosf(st);
}

// ---------------------------------------------------------------------------
// Graph attention over the first 36 timesteps (per batch element).
// S = Q K^T / 6, row softmax -> alpha; h[:36, b, :36] += alpha @ V.
// Also writes a^T (B x 1296) for the pairwise-distance term.
// ---------------------------------------------------------------------------
__global__ __launch_bounds__(128) void graph_kernel(
    const float* __restrict__ q36, const float* __restrict__ k36,
    const float* __restrict__ v36, float* __restrict__ h,
    float* __restrict__ aT) {
  const int b = blockIdx.x;
  __shared__ float Q[36][37], Kk[36][37], V[36][37], P[36][36];
  for (int idx = threadIdx.x; idx < 36 * 36; idx += blockDim.x) {
    int t = idx / 36, d = idx - t * 36;
    size_t off = (size_t)t * BATCH * 36 + (size_t)b * 36 + d;
    Q[t][d] = q36[off];
    Kk[t][d] = k36[off];
    V[t][d] = v36[off];
  }
  __syncthreads();
  if (threadIdx.x < 36) {
    const int dst = threadIdx.x;
    float srow[36];
    float mx = -__builtin_inff();
    for (int src = 0; src < 36; ++src) {
      float s = 0.f;
      for (int d = 0; d < 36; ++d) s += Q[dst][d] * Kk[src][d];
      s *= (1.0f / 6.0f);
      srow[src] = s;
      mx = fmaxf(mx, s);
    }
    float sum = 0.f;
    for (int src = 0; src < 36; ++src) {
      float e = expf(srow[src] - mx);
      srow[src] = e;
      sum += e;
    }
    float inv = 1.0f / sum;
    for (int src = 0; src < 36; ++src) {
      float al = srow[src] * inv;
      P[dst][src] = al;
      aT[(size_t)b * 1296 + src * 36 + dst] = al;  // edge e = src*36 + dst
    }
  }
  __syncthreads();
  for (int idx = threadIdx.x; idx < 1296; idx += blockDim.x) {
    int dst = idx / 36, d = idx - dst * 36;
    float acc = 0.f;
    for (int src = 0; src < 36; ++src) acc += P[dst][src] * V[src][d];
    h[(size_t)dst * BATCH * 72 + (size_t)b * 72 + d] += acc;
  }
}

// ---------------------------------------------------------------------------
// Fused flash-attention MHA for one (batch, head). qkv: (T,B,216) packed.
// Q/K/V (215x18 zero-padded to 224x20) in LDS; per wave: 16 query rows,
// online softmax over 14 key tiles. S and P*V via fp32 WMMA. All inner-loop
// LDS reads are unconditional (padding columns are zero / clamped-benign).
// ---------------------------------------------------------------------------
__global__ __launch_bounds__(256) void mha_kernel(
    const float* __restrict__ qkv, const int* __restrict__ lengths,
    float* __restrict__ o) {
  const int b = blockIdx.x >> 2;
  const int hh = blockIdx.x & 3;
  __shared__ float Qs[T_PAD][20], Ks[T_PAD][20], Vs[T_PAD][20];
  __shared__ float Ps[8][16][17];

  const int tid = threadIdx.x;
  const int B216 = BATCH * 216;
  for (int idx = tid; idx < T_PAD * 20; idx += 256) {
    int t = idx / 20, d = idx - t * 20;
    float q = 0.f, k = 0.f, v = 0.f;
    if (t < T_LEN && d < 18) {
      const float* base = qkv + (size_t)t * B216 + (size_t)b * 216;
      q = base[hh * 18 + d];
      k = base[72 + hh * 18 + d];
      v = base[144 + hh * 18 + d];
    }
    Qs[t][d] = q;
    Ks[t][d] = k;
    Vs[t][d] = v;
  }
  __syncthreads();

  const int lane = tid & 31, wv = tid >> 5;
  const int half = lane >> 4, l15 = lane & 15;
  const int len = lengths[b];
  const float rscale = rsqrtf(18.0f);
  const size_t B72 = (size_t)BATCH * 72;

  for (int qt = wv; qt < 14; qt += 8) {
    const int qbase = qt * 16;
    float mrow[8], lrow[8];
    v8f O0{}, O1{};
#pragma unroll
    for (int j = 0; j < 8; ++j) {
      mrow[j] = -__builtin_inff();
      lrow[j] = 0.f;
    }
    for (int kt = 0; kt < 14; ++kt) {
      const int kbase = kt * 16;
      v8f S{};
#pragma unroll
      for (int k0 = 0; k0 < 20; k0 += 4) {  // K = 18, zero-padded to 20
        int ka = k0 + 2 * half;
        v2f a, bb;
        a.x = Qs[qbase + l15][ka];
        a.y = Qs[qbase + l15][ka + 1];
        bb.x = Ks[kbase + l15][ka];
        bb.y = Ks[kbase + l15][ka + 1];
        S = wmma4(a, bb, S);
      }
      const int key = kbase + l15;  // column index shared by all 8 rows
#pragma unroll
      for (int j = 0; j < 8; ++j) {
        float s = S[j] * rscale;
        if (key >= T_LEN) s = -__builtin_inff();   // padding keys: weight 0
        else if (key >= len) s = -1e9f;            // reference mask value
        S[j] = s;
      }
      // online softmax; rows live per-VGPR, striped over each 16-lane half
#pragma unroll
      for (int j = 0; j < 8; ++j) {
        float rm = S[j];
        for (int off = 1; off < 16; off <<= 1)
          rm = fmaxf(rm, __shfl_xor(rm, off, 32));
        float mn = fmaxf(mrow[j], rm);
        float sc = expf(mrow[j] - mn);  // first tile: exp(-inf)=0
        float p = expf(S[j] - mn);
        float rs = p;
        for (int off = 1; off < 16; off <<= 1) rs += __shfl_xor(rs, off, 32);
        lrow[j] = lrow[j] * sc + rs;
        O0[j] *= sc;
        O1[j] *= sc;
        mrow[j] = mn;
        Ps[wv][j + 8 * half][l15] = p;  // C-layout -> LDS for A-fragment reads
      }
      __builtin_amdgcn_wave_barrier();  // keep LDS store->load ordering
      // O += P (16x16) @ V_tile (16x18): N-tiles d[0..15] and d[16..17].
      // Second tile uses a clamped column (16 + (l15&1)): lanes 2..15 compute
      // benign finite values into O1 columns that are never stored.
      const int d1 = 16 + (l15 & 1);
#pragma unroll
      for (int k0 = 0; k0 < 16; k0 += 4) {
        int ka = k0 + 2 * half;
        v2f a, b0, b1;
        a.x = Ps[wv][l15][ka];
        a.y = Ps[wv][l15][ka + 1];
        b0.x = Vs[kbase + ka][l15];
        b0.y = Vs[kbase + ka + 1][l15];
        O0 = wmma4(a, b0, O0);
        b1.x = Vs[kbase + ka][d1];
        b1.y = Vs[kbase + ka + 1][d1];
        O1 = wmma4(a, b1, O1);
      }
      __builtin_amdgcn_wave_barrier();
    }
#pragma unroll
    for (int j = 0; j < 8; ++j) {
      int t = qbase + j + 8 * half;
      if (t < T_LEN) {
        float inv = 1.0f / lrow[j];
        o[(size_t)t * B72 + (size_t)b * 72 + hh * 18 + l15] = O0[j] * inv;
        if (l15 < 2)
          o[(size_t)t * B72 + (size_t)b * 72 + hh * 18 + 16 + l15] = O1[j] * inv;
      }
    }
  }
}

// ---------------------------------------------------------------------------
// h = LayerNorm(h + t) * g + b   (one wave per 72-wide row)
// ---------------------------------------------------------------------------
__global__ __launch_bounds__(256) void addln_kernel(float* __restrict__ h,
                                                    const float* __restrict__ t,
                                                    const float* __restrict__ g,
                                                    const float* __restrict__ be) {
  int row = blockIdx.x * 8 + (threadIdx.x >> 5);
  if (row >= TBROWS) return;
  int lane = threadIdx.x & 31;
  size_t base = (size_t)row * 72;
  float z0 = h[base + lane] + t[base + lane];
  float z1 = h[base + 32 + lane] + t[base + 32 + lane];
  float z2 = (lane < 8) ? (h[base + 64 + lane] + t[base + 64 + lane]) : 0.f;
  float s = z0 + z1 + z2;
  float ss = z0 * z0 + z1 * z1 + z2 * z2;
  for (int off = 1; off < 32; off <<= 1) {
    s += __shfl_xor(s, off, 32);
    ss += __shfl_xor(ss, off, 32);
  }
  float mean = s * (1.0f / 72.0f);
  float var = ss * (1.0f / 72.0f) - mean * mean;
  float inv = rsqrtf(var + 1e-5f);
  h[base + lane] = (z0 - mean) * inv * g[lane] + be[lane];
  h[base + 32 + lane] = (z1 - mean) * inv * g[32 + lane] + be[32 + lane];
  if (lane < 8)
    h[base + 64 + lane] = (z2 - mean) * inv * g[64 + lane] + be[64 + lane];
}

// ---------------------------------------------------------------------------
// Masked mean pool: pc[b, c] = sum_{t<len[b]} h[t,b,c] / (len[b]+1)
// ---------------------------------------------------------------------------
__global__ void pool_kernel(const float* __restrict__ h,
                            const int* __restrict__ lengths,
                            float* __restrict__ pc) {
  int idx = blockIdx.x * blockDim.x + threadIdx.x;
  if (idx >= BATCH * 72) return;
  int b = idx / 72, c = idx - b * 72;
  int len = lengths[b];
  float s = 0.f;
  for (int t = 0; t < len; ++t)
    s += h[(size_t)t * BATCH * 72 + (size_t)b * 72 + c];
  pc[(size_t)b * 108 + c] = s / ((float)len + 1.0f);
}

// ---------------------------------------------------------------------------
// distance = mean_{i,j} sqrt(max(0, G[i,i]+G[j,j]-2 G[i,j])), G = a a^T
// Two-stage deterministic reduction (no float atomics -> replay-stable).
// ---------------------------------------------------------------------------
__global__ __launch_bounds__(128) void dist_row_kernel(
    const float* __restrict__ G, float* __restrict__ part) {
  int i = blockIdx.x;
  float sqi = G[(size_t)i * 512 + i];
  float s = 0.f;
  for (int j = threadIdx.x; j < 512; j += 128) {
    float d2 = sqi + G[(size_t)j * 512 + j] - 2.0f * G[(size_t)i * 512 + j];
    s += sqrtf(fmaxf(d2, 0.f));
  }
  __shared__ float red[128];
  red[threadIdx.x] = s;
  __syncthreads();
  for (int o = 64; o > 0; o >>= 1) {
    if (threadIdx.x < o) red[threadIdx.x] += red[threadIdx.x + o];
    __syncthreads();
  }
  if (threadIdx.x == 0) part[i] = red[0];
}

__global__ __launch_bounds__(256) void dist_final_kernel(
    const float* __restrict__ part, float* __restrict__ out) {
  __shared__ float red[256];
  red[threadIdx.x] = part[threadIdx.x] + part[threadIdx.x + 256];
  __syncthreads();
  for (int o = 128; o > 0; o >>= 1) {
    if (threadIdx.x < o) red[threadIdx.x] += red[threadIdx.x + o];
    __syncthreads();
  }
  if (threadIdx.x == 0) out[0] = red[0] * (1.0f / (512.0f * 512.0f));
}

// ---------------------------------------------------------------------------
// Host launcher. d_in order = jax pytree flatten of setup_inputs():
//   0:src 1:static 2:times 3:lengths, then params dict (sorted keys):
//   4:emb_b 5:emb_w 6:enc_b 7:enc_w 8:k_b 9:k_w
//   10..21: layer0 {in_b,in_w,lin1_b,lin1_w,lin2_b,lin2_w,ln1_b,ln1_g,ln2_b,ln2_g,out_b,out_w}
//   22..33: layer1 (same), 34:mlp1_b 35:mlp1_w 36:mlp2_b 37:mlp2_w
//   38:q_b 39:q_w 40:skip_b 41:skip_w 42:v_b 43:v_w
// ---------------------------------------------------------------------------
static inline dim3 gemm_grid(int M, int N) {
  return dim3((unsigned)((N + 15) / 16), (unsigned)((M + 63) / 64));
}

extern "C" void kernel_launch(void* const* d_in, const int* in_sizes, int n_in,
                              void* d_out, int out_size, void* d_ws,
                              size_t ws_size, hipStream_t stream) {
  (void)in_sizes; (void)n_in; (void)out_size; (void)ws_size;
  const float* src = (const float*)d_in[0];
  const float* statx = (const float*)d_in[1];
  const float* times = (const float*)d_in[2];
  const int* lens = (const int*)d_in[3];
  const float* emb_b = (const float*)d_in[4];
  const float* emb_w = (const float*)d_in[5];
  const float* enc_b = (const float*)d_in[6];
  const float* enc_w = (const float*)d_in[7];
  const float* k_b = (const float*)d_in[8];
  const float* k_w = (const float*)d_in[9];
  const float* mlp1_b = (const float*)d_in[34];
  const float* mlp1_w = (const float*)d_in[35];
  const float* mlp2_b = (const float*)d_in[36];
  const float* mlp2_w = (const float*)d_in[37];
  const float* q_b = (const float*)d_in[38];
  const float* q_w = (const float*)d_in[39];
  const float* skip_b = (const float*)d_in[40];
  const float* skip_w = (const float*)d_in[41];
  const float* v_b = (const float*)d_in[42];
  const float* v_w = (const float*)d_in[43];

  // workspace arena (floats)
  float* w = (float*)d_ws;
  size_t off = 0;
  float* x = w + off;    off += (size_t)TBROWS * 36;
  float* h = w + off;    off += (size_t)TBROWS * 72;
  float* big = w + off;  off += (size_t)TBROWS * 216;  // qkv / tmp / ff hidden
  float* obuf = w + off; off += (size_t)TBROWS * 72;   // attn out / ff out
  float* q36 = w + off;  off += (size_t)36 * BATCH * 36;
  float* k36 = w + off;  off += (size_t)36 * BATCH * 36;
  float* v36 = w + off;  off += (size_t)36 * BATCH * 36;
  float* aT = w + off;   off += (size_t)BATCH * 1296;
  float* G = w + off;    off += (size_t)BATCH * BATCH;
  float* pc = w + off;   off += (size_t)BATCH * 108;   // [pooled | emb]
  float* hid = w + off;  off += (size_t)BATCH * 108;
  float* part = w + off; off += 512;

  const dim3 blk(128);
  // x = (src[:,:,:36] @ enc_w + enc_b) * sqrt(36)
  gemm_wmma_kernel<36><<<gemm_grid(TBROWS, 36), blk, 0, stream>>>(
      src, 72, enc_w, 36, 0, enc_b, x, 36, 0, TBROWS, 36, 6.0f, 0);
  // q/k/v only needed for t < 36 (edges index the time axis 0..35)
  gemm_wmma_kernel<36><<<gemm_grid(36 * BATCH, 36), blk, 0, stream>>>(
      x, 36, q_w, 36, 0, q_b, q36, 36, 0, 36 * BATCH, 36, 1.0f, 0);
  gemm_wmma_kernel<36><<<gemm_grid(36 * BATCH, 36), blk, 0, stream>>>(
      x, 36, k_w, 36, 0, k_b, k36, 36, 0, 36 * BATCH, 36, 1.0f, 0);
  gemm_wmma_kernel<36><<<gemm_grid(36 * BATCH, 36), blk, 0, stream>>>(
      x, 36, v_w, 36, 0, v_b, v36, 36, 0, 36 * BATCH, 36, 1.0f, 0);
  // h[:, :, 0:36] = x @ skip_w + skip_b (graph kernel adds agg on top)
  gemm_wmma_kernel<36><<<gemm_grid(TBROWS, 36), blk, 0, stream>>>(
      x, 36, skip_w, 36, 0, skip_b, h, 72, 0, TBROWS, 36, 1.0f, 0);
  // h[:, :, 36:72] = positional encoding
  pe_kernel<<<(TBROWS * 18 + 255) / 256, 256, 0, stream>>>(times, h);
  // graph attention (adds agg into h[:36]), writes alpha^T
  graph_kernel<<<BATCH, 128, 0, stream>>>(q36, k36, v36, h, aT);
  // emb -> pc[:, 72:108]
  gemm_wmma_kernel<9><<<gemm_grid(BATCH, 36), blk, 0, stream>>>(
      statx, 9, emb_w, 36, 0, emb_b, pc, 108, 72, BATCH, 36, 1.0f, 0);
  // G = a @ a^T  (512 x 512, K = 1296), then distance
  gemm_wmma_kernel<1296><<<gemm_grid(BATCH, BATCH), blk, 0, stream>>>(
      aT, 1296, aT, 1296, 1, nullptr, G, 512, 0, 512, 512, 1.0f, 0);
  dist_row_kernel<<<512, 128, 0, stream>>>(G, part);
  dist_final_kernel<<<1, 256, 0, stream>>>(part, (float*)d_out + 1024);

  for (int l = 0; l < 2; ++l) {
    const int base = 10 + 12 * l;
    const float* in_b = (const float*)d_in[base + 0];
    const float* in_w = (const float*)d_in[base + 1];
    const float* lin1_b = (const float*)d_in[base + 2];
    const float* lin1_w = (const float*)d_in[base + 3];
    const float* lin2_b = (const float*)d_in[base + 4];
    const float* lin2_w = (const float*)d_in[base + 5];
    const float* ln1_b = (const float*)d_in[base + 6];
    const float* ln1_g = (const float*)d_in[base + 7];
    const float* ln2_b = (const float*)d_in[base + 8];
    const float* ln2_g = (const float*)d_in[base + 9];
    const float* out_b = (const float*)d_in[base + 10];
    const float* out_w = (const float*)d_in[base + 11];

    gemm_wmma_kernel<72><<<gemm_grid(TBROWS, 216), blk, 0, stream>>>(
        h, 72, in_w, 216, 0, in_b, big, 216, 0, TBROWS, 216, 1.0f, 0);
    mha_kernel<<<BATCH * 4, 256, 0, stream>>>(big, lens, obuf);
    gemm_wmma_kernel<72><<<gemm_grid(TBROWS, 72), blk, 0, stream>>>(
        obuf, 72, out_w, 72, 0, out_b, big, 72, 0, TBROWS, 72, 1.0f, 0);
    addln_kernel<<<(TBROWS + 7) / 8, 256, 0, stream>>>(h, big, ln1_g, ln1_b);
    gemm_wmma_kernel<72><<<gemm_grid(TBROWS, 128), blk, 0, stream>>>(
        h, 72, lin1_w, 128, 0, lin1_b, big, 128, 0, TBROWS, 128, 1.0f, 1);
    gemm_wmma_kernel<128><<<gemm_grid(TBROWS, 72), blk, 0, stream>>>(
        big, 128, lin2_w, 72, 0, lin2_b, obuf, 72, 0, TBROWS, 72, 1.0f, 0);
    addln_kernel<<<(TBROWS + 7) / 8, 256, 0, stream>>>(h, obuf, ln2_g, ln2_b);
  }

  pool_kernel<<<(BATCH * 72 + 255) / 256, 256, 0, stream>>>(h, lens, pc);
  gemm_wmma_kernel<108><<<gemm_grid(BATCH, 108), blk, 0, stream>>>(
      pc, 108, mlp1_w, 108, 0, mlp1_b, hid, 108, 0, BATCH, 108, 1.0f, 1);
  gemm_wmma_kernel<108><<<gemm_grid(BATCH, 2), blk, 0, stream>>>(
      hid, 108, mlp2_w, 2, 0, mlp2_b, (float*)d_out, 2, 0, BATCH, 2, 1.0f, 0);
}